// MulRel_66133906424098
// MI455X (gfx1250) — compile-verified
//
#include <hip/hip_runtime.h>
#include <hip/hip_bf16.h>

typedef float v2f __attribute__((ext_vector_type(2)));
typedef float v4f __attribute__((ext_vector_type(4)));
typedef float v8f __attribute__((ext_vector_type(8)));

#define DIM 512
#define RREL 8
#define BATCH 16
#define SEQ 512
#define STR (DIM + 4)      // padded LDS row stride (516 % 64 == 4 -> conflict-free)
#define NEG_BIG (-1e10f)

// ---------------------------------------------------------------------------
// Kernel A: row-wise L2 normalize w [8,512]; c[r] = sum(w_norm^2) (== 1.0)
// One block, 8 waves, wave r handles row r.
// ---------------------------------------------------------------------------
__global__ __launch_bounds__(256) void normalize_w_kernel(
    const float* __restrict__ w, float* __restrict__ wn, float* __restrict__ c) {
  const int wave = threadIdx.x >> 5;   // row r
  const int lane = threadIdx.x & 31;
  float s = 0.f;
  for (int k = lane; k < DIM; k += 32) {
    float v = w[wave * DIM + k];
    s += v * v;
  }
  #pragma unroll
  for (int off = 16; off; off >>= 1) s += __shfl_xor(s, off, 32);
  float n = sqrtf(s);
  float denom = fmaxf(n, 1e-12f);
  float inv = 1.0f / denom;
  for (int k = lane; k < DIM; k += 32) wn[wave * DIM + k] = w[wave * DIM + k] * inv;
  if (lane == 0) c[wave] = s * inv * inv;
}

// ---------------------------------------------------------------------------
// Kernel B: projection GEMM via V_WMMA_F32_16X16X4_F32.
// Y[b,r,s] = sum_d X[b*rows+s, d] * wn[r, d]
// Block: 256 thr (8 waves) handles a 16-row tile; K=512 split across waves
// (64 each = 16 chained WMMAs); partial 16x16 C tiles reduced through LDS.
// ---------------------------------------------------------------------------
__global__ __launch_bounds__(256) void proj_wmma_kernel(
    const float* __restrict__ X,    // [n_rows, DIM]
    const float* __restrict__ wn,   // [8, DIM]
    float* __restrict__ Yt,         // [B, 8, rows_per_batch]
    int rows_per_batch) {
  __shared__ float xs[16 * STR];      // 33,024 B
  __shared__ float wsh[8 * STR];      // 16,512 B
  __shared__ float part[8 * 256];     //  8,192 B
  const int tid = threadIdx.x;
  const int row_base = blockIdx.x * 16;

  for (int i = tid; i < RREL * DIM; i += 256) {
    int r = i >> 9, k = i & (DIM - 1);
    wsh[r * STR + k] = wn[i];
  }
  for (int i = tid; i < 16 * DIM; i += 256) {
    int r = i >> 9, k = i & (DIM - 1);
    xs[r * STR + k] = X[(size_t)(row_base + r) * DIM + k];
  }
  __syncthreads();

  const int wave = tid >> 5;
  const int lane = tid & 31;
  const int half = lane >> 4;         // 0: K={k0,k0+1}, 1: K={k0+2,k0+3}
  const int m    = lane & 15;         // A row / B col
  const int kbase = wave * 64;

  v8f acc = {};
  #pragma unroll
  for (int step = 0; step < 16; ++step) {
    const int k0 = kbase + step * 4 + half * 2;
    v2f a, bfrag;
    a.x = xs[m * STR + k0];
    a.y = xs[m * STR + k0 + 1];
    float b0 = wsh[(m & 7) * STR + k0];
    float b1 = wsh[(m & 7) * STR + k0 + 1];
    bfrag.x = (m < RREL) ? b0 : 0.0f;   // pad N=8..15 with zero columns
    bfrag.y = (m < RREL) ? b1 : 0.0f;
    acc = __builtin_amdgcn_wmma_f32_16x16x4_f32(
        /*neg_a=*/false, a, /*neg_b=*/false, bfrag,
        /*c_mod=*/(short)0, acc, /*reuse_a=*/false, /*reuse_b=*/false);
  }

  #pragma unroll
  for (int v = 0; v < 8; ++v) part[wave * 256 + v * 32 + lane] = acc[v];
  __syncthreads();

  // thread t reduces C element (vgpr = t>>5, lane = t&31) across 8 waves
  float sum = 0.f;
  #pragma unroll
  for (int wv = 0; wv < 8; ++wv) sum += part[wv * 256 + tid];
  const int v = tid >> 5;
  const int l = tid & 31;
  const int M = v + 8 * (l >> 4);     // C/D layout: VGPR v, lanes 16-31 -> M+8
  const int N = l & 15;
  if (N < RREL) {
    const int row = row_base + M;
    const int b = row / rows_per_batch;
    const int s = row - b * rows_per_batch;
    Yt[((size_t)b * RREL + N) * rows_per_batch + s] = sum;
  }
}

// ---------------------------------------------------------------------------
// Kernel C: per (b,r,s) masked softmax over t and weighted sum of tar_wt.
// score_t = a*tw[t] + (1 - sm*tm[t])*NEG_BIG,  a = src_wt[b,r,s]*c_r.
// Lane-parallel over t (wave32 shfl_xor reductions); each wave handles 8 s.
// Block = (b, r, 64-s chunk); tar row + NEG_BIG-scaled tar_mask in LDS.
// ---------------------------------------------------------------------------
__global__ __launch_bounds__(256) void attn_g_kernel(
    const float* __restrict__ src_wt,   // [B,8,S]
    const float* __restrict__ tar_wt,   // [B,8,T]
    const float* __restrict__ src_mask, // [B,S]
    const float* __restrict__ tar_mask, // [B,T]
    const float* __restrict__ c,        // [8]
    float* __restrict__ g) {            // [B,8,S]
  __shared__ float tw[SEQ];
  __shared__ float tmn[SEQ];            // tar_mask[t] * NEG_BIG
  const int tid  = threadIdx.x;
  const int wave = tid >> 5;
  const int lane = tid & 31;
  const int chunk = blockIdx.x & 7;            // 64-wide s chunk
  const int r     = (blockIdx.x >> 3) & 7;
  const int b     = blockIdx.x >> 6;

  const float* twp = tar_wt + ((size_t)b * RREL + r) * SEQ;
  const float* tmp = tar_mask + (size_t)b * SEQ;
  for (int i = tid; i < SEQ; i += 256) {
    tw[i]  = twp[i];
    tmn[i] = tmp[i] * NEG_BIG;
  }
  __syncthreads();

  const int s0 = chunk * 64 + wave * 8;        // this wave's 8 s-values
  float a_l = 0.f, sm_l = 0.f;
  if (lane < 8) {
    a_l  = src_wt[((size_t)b * RREL + r) * SEQ + s0 + lane] * c[r];
    sm_l = src_mask[(size_t)b * SEQ + s0 + lane];
  }

  #pragma unroll
  for (int i = 0; i < 8; ++i) {
    const float a  = __shfl(a_l, i, 32);
    const float sm = __shfl(sm_l, i, 32);
    // bias_t = (1 - sm*tm[t])*NEG_BIG = NEG_BIG - sm*(tm[t]*NEG_BIG)
    float mx = -3.0e38f;
    for (int t = lane; t < SEQ; t += 32) {
      float sc = fmaf(a, tw[t], fmaf(-sm, tmn[t], NEG_BIG));
      mx = fmaxf(mx, sc);
    }
    #pragma unroll
    for (int off = 16; off; off >>= 1) mx = fmaxf(mx, __shfl_xor(mx, off, 32));

    float se = 0.f, gs = 0.f;
    for (int t = lane; t < SEQ; t += 32) {
      float sc = fmaf(a, tw[t], fmaf(-sm, tmn[t], NEG_BIG));
      float e = __expf(sc - mx);
      se += e;
      gs = fmaf(e, tw[t], gs);
    }
    #pragma unroll
    for (int off = 16; off; off >>= 1) {
      se += __shfl_xor(se, off, 32);
      gs += __shfl_xor(gs, off, 32);
    }
    if (lane == 0) g[((size_t)b * RREL + r) * SEQ + s0 + i] = gs / se;
  }
}

// ---------------------------------------------------------------------------
// Kernel D: out[b,s,d] = sum_r g[b,r,s] * wn[r,d]   (float4 per thread)
// ---------------------------------------------------------------------------
__global__ __launch_bounds__(256) void out_proj_kernel(
    const float* __restrict__ g,    // [B,8,S]
    const float* __restrict__ wn,   // [8,DIM]
    float* __restrict__ out) {      // [B,S,DIM]
  __shared__ float wsh[RREL * DIM];   // 16 KB
  const int tid = threadIdx.x;
  for (int i = tid; i < RREL * DIM; i += 256) wsh[i] = wn[i];
  __syncthreads();

  const size_t idx = (size_t)blockIdx.x * 256 + tid;  // one float4 of out
  const int d4 = (int)(idx & 127);          // DIM/4 = 128
  const int s  = (int)((idx >> 7) & 511);
  const int b  = (int)(idx >> 16);

  v4f acc = {};
  #pragma unroll
  for (int r = 0; r < RREL; ++r) {
    const float gv = g[((size_t)b * RREL + r) * SEQ + s];
    const v4f wv = *(const v4f*)&wsh[r * DIM + d4 * 4];
    acc += gv * wv;
  }
  *(v4f*)&out[idx * 4] = acc;
}

// ---------------------------------------------------------------------------
extern "C" void kernel_launch(void* const* d_in, const int* in_sizes, int n_in,
                              void* d_out, int out_size, void* d_ws, size_t ws_size,
                              hipStream_t stream) {
  const float* src_emb  = (const float*)d_in[0];  // [16,512,512]
  const float* tar_emb  = (const float*)d_in[1];  // [16,512,512]
  const float* src_mask = (const float*)d_in[2];  // [16,512]
  const float* tar_mask = (const float*)d_in[3];  // [16,512]
  const float* msw      = (const float*)d_in[4];  // [8,512]
  float* out = (float*)d_out;                     // [16,512,512]

  // workspace layout (floats)
  float* ws  = (float*)d_ws;
  float* wn  = ws;                    // 4096
  float* c   = ws + 4096;             // 8 (+pad)
  float* swt = ws + 4352;             // 65536  [B,8,S]
  float* twt = swt + BATCH * RREL * SEQ;  // 65536  [B,8,T]
  float* gb  = twt + BATCH * RREL * SEQ;  // 65536  [B,8,S]

  normalize_w_kernel<<<1, 256, 0, stream>>>(msw, wn, c);

  const int n_row_tiles = (BATCH * SEQ) / 16;   // 512
  proj_wmma_kernel<<<n_row_tiles, 256, 0, stream>>>(src_emb, wn, swt, SEQ);
  proj_wmma_kernel<<<n_row_tiles, 256, 0, stream>>>(tar_emb, wn, twt, SEQ);

  attn_g_kernel<<<BATCH * RREL * (SEQ / 64), 256, 0, stream>>>(
      swt, twt, src_mask, tar_mask, c, gb);

  out_proj_kernel<<<(BATCH * SEQ * (DIM / 4)) / 256, 256, 0, stream>>>(gb, wn, out);
}